// InvariantPointAttention_66984309948711
// MI455X (gfx1250) — compile-verified
//
#include <hip/hip_runtime.h>
#include <hip/hip_bf16.h>
#include <math.h>

// ---------------------------------------------------------------------------
// InvariantPointAttention fused kernel set for MI455X (gfx1250, wave32, WMMA)
//
// Roofline: z is [1024,1024,128] f32 = 512 MB and dominates all traffic.
// We stream z exactly once from HBM: the bias_z GEMM (z @ Wz) and the
// o1 contraction (att * z) are fused into one pass over 16x16 (i,j) chunks;
// the second use of each 128KB chunk hits L2 (192 MB). ~550 MB total HBM
// => ~24 us at 23.3 TB/s. All large contractions use v_wmma_f32_16x16x4_f32.
// k/v staging overlaps the bias WMMA via GLOBAL_LOAD_ASYNC_TO_LDS_B128
// (ASYNCcnt-tracked DMA), retired with s_wait_asynccnt before the consumer
// barrier.
// ---------------------------------------------------------------------------

typedef float v2f __attribute__((ext_vector_type(2)));
typedef float v8f __attribute__((ext_vector_type(8)));

#define WMMA_F32X4(a, b, c) \
    __builtin_amdgcn_wmma_f32_16x16x4_f32(false, (a), false, (b), (short)0, (c), false, false)

#define LL 1024
#define DIN 384
#define DZ 128
#define C_SQU 0.25f                    /* 1/sqrt(16) */
#define WGT_SCALE 0.11785113019775792f /* 0.5 * sqrt(2/(9*4)) */

__device__ inline v8f vzero8() {
    v8f v;
#pragma unroll
    for (int i = 0; i < 8; ++i) v[i] = 0.f;
    return v;
}

// Async DMA: global -> LDS, 16 bytes per lane, tracked by ASYNCcnt.
// Generic LDS pointers carry the byte offset in their low 32 bits.
__device__ inline void async_cp16(float* lds_dst, const float* gsrc) {
    unsigned lds = (unsigned)(uintptr_t)lds_dst;
    unsigned long long ga = (unsigned long long)(uintptr_t)gsrc;
    asm volatile("global_load_async_to_lds_b128 %0, %1, off"
                 :: "v"(lds), "v"(ga) : "memory");
}
__device__ inline void wait_async0() {
    asm volatile("s_wait_asynccnt 0x0" ::: "memory");
}

// ---------------------------------------------------------------------------
// Zero-fill (atomic accumulators must start at 0; harness poisons ws)
// ---------------------------------------------------------------------------
__global__ void ipa_zero_kernel(float* __restrict__ p, int n) {
    int i = blockIdx.x * blockDim.x + threadIdx.x;
    if (i < n) p[i] = 0.f;
}

// ---------------------------------------------------------------------------
// Projections: q1/k1/v1 (L x H*C), transformed q2/k2 points (+ squared norms),
// transformed v3 points.  One block per residue row.  ~0.6 GFLOP total: VALU.
// ---------------------------------------------------------------------------
__global__ __launch_bounds__(128) void ipa_proj_kernel(
    const float* __restrict__ s, const float* __restrict__ rot,
    const float* __restrict__ trans,
    const float* __restrict__ Wq1, const float* __restrict__ Wk1,
    const float* __restrict__ Wv1, const float* __restrict__ Wq2,
    const float* __restrict__ Wk2, const float* __restrict__ Wv3,
    float* __restrict__ q1b, float* __restrict__ k1b, float* __restrict__ v1b,
    float* __restrict__ qpb, float* __restrict__ kpb,
    float* __restrict__ q2nb, float* __restrict__ k2nb,
    float* __restrict__ v3b) {
    __shared__ float srow[DIN];
    __shared__ float sraw[144];
    __shared__ float str[96];
    const int i = blockIdx.x;
    const int t = threadIdx.x;

    for (int k = t; k < DIN; k += 128) srow[k] = s[i * DIN + k];
    __syncthreads();

    // q1 / k1 / v1: column t of the 128-wide projections
    {
        float aq = 0.f, ak = 0.f, av = 0.f;
        for (int k = 0; k < DIN; ++k) {
            float sv = srow[k];
            aq = fmaf(sv, Wq1[k * 128 + t], aq);
            ak = fmaf(sv, Wk1[k * 128 + t], ak);
            av = fmaf(sv, Wv1[k * 128 + t], av);
        }
        q1b[i * 128 + t] = aq;
        k1b[i * 128 + t] = ak;
        v1b[i * 128 + t] = av;
    }

    float R[9], T[3];
#pragma unroll
    for (int a = 0; a < 9; ++a) R[a] = rot[i * 9 + a];
#pragma unroll
    for (int a = 0; a < 3; ++a) T[a] = trans[i * 3 + a];

    // ---- q2: raw projection, rigid transform, squared norm ----
    __syncthreads();
    if (t < 96) {
        float a = 0.f;
        for (int k = 0; k < DIN; ++k) a = fmaf(srow[k], Wq2[k * 96 + t], a);
        sraw[t] = a;
    }
    __syncthreads();
    if (t < 32) {  // (h,q) pair
        int base = t * 3;
        float x = sraw[base], y = sraw[base + 1], zz = sraw[base + 2];
#pragma unroll
        for (int d = 0; d < 3; ++d) {
            float v = x * R[0 * 3 + d] + y * R[1 * 3 + d] + zz * R[2 * 3 + d] + T[d];
            str[base + d] = v;
            qpb[i * 96 + base + d] = v;
        }
    }
    __syncthreads();
    if (t < 8) {
        float s2 = 0.f;
        for (int a = 0; a < 12; ++a) { float v = str[t * 12 + a]; s2 = fmaf(v, v, s2); }
        q2nb[i * 8 + t] = s2;
    }
    __syncthreads();

    // ---- k2 ----
    if (t < 96) {
        float a = 0.f;
        for (int k = 0; k < DIN; ++k) a = fmaf(srow[k], Wk2[k * 96 + t], a);
        sraw[t] = a;
    }
    __syncthreads();
    if (t < 32) {
        int base = t * 3;
        float x = sraw[base], y = sraw[base + 1], zz = sraw[base + 2];
#pragma unroll
        for (int d = 0; d < 3; ++d) {
            float v = x * R[0 * 3 + d] + y * R[1 * 3 + d] + zz * R[2 * 3 + d] + T[d];
            str[base + d] = v;
            kpb[i * 96 + base + d] = v;
        }
    }
    __syncthreads();
    if (t < 8) {
        float s2 = 0.f;
        for (int a = 0; a < 12; ++a) { float v = str[t * 12 + a]; s2 = fmaf(v, v, s2); }
        k2nb[i * 8 + t] = s2;
    }
    __syncthreads();

    // ---- v3: 144 columns, transformed points ----
    for (int col = t; col < 144; col += 128) {
        float a = 0.f;
        for (int k = 0; k < DIN; ++k) a = fmaf(srow[k], Wv3[k * 144 + col], a);
        sraw[col] = a;
    }
    __syncthreads();
    if (t < 48) {  // (h,p) pair
        int base = t * 3;
        float x = sraw[base], y = sraw[base + 1], zz = sraw[base + 2];
#pragma unroll
        for (int d = 0; d < 3; ++d)
            v3b[i * 144 + base + d] =
                x * R[0 * 3 + d] + y * R[1 * 3 + d] + zz * R[2 * 3 + d] + T[d];
    }
}

// ---------------------------------------------------------------------------
// Concatenate W1|W2|W3 -> Wcat [1296][384], bsum = b1+b2+b3
// ---------------------------------------------------------------------------
__global__ void ipa_wcat_kernel(const float* __restrict__ W1, const float* __restrict__ b1,
                                const float* __restrict__ W2, const float* __restrict__ b2,
                                const float* __restrict__ W3, const float* __restrict__ b3,
                                float* __restrict__ Wcat, float* __restrict__ bsum) {
    int idx = blockIdx.x * blockDim.x + threadIdx.x;
    if (idx < 1296 * 384) {
        int row = idx / 384, col = idx - row * 384;
        float v;
        if (row < 1024) v = W1[row * 384 + col];
        else if (row < 1152) v = W2[(row - 1024) * 384 + col];
        else v = W3[(row - 1152) * 384 + col];
        Wcat[idx] = v;
    }
    if (idx < 384) bsum[idx] = b1[idx] + b2[idx] + b3[idx];
}

// ---------------------------------------------------------------------------
// Fused attention core.
// grid = (64 i-tiles, 16 j-segments), block = 256 (8 waves of 32).
// Per 16x16 (i,j) chunk:
//   async-DMA k/v staging to LDS (overlaps step1)
//   step1: bias_z = z @ Wz via v_wmma_f32_16x16x4_f32 (A = z rows from HBM)
//   step2: + q1.k1 * C_SQU - softplus(gama)*W_C/2 * ||q2-k2||^2   (VALU)
//   step3: o1 += att * z via WMMA (z re-read hits L2); o2/o3 h-batched VALU
// ---------------------------------------------------------------------------
__global__ __launch_bounds__(256) void ipa_att_kernel(
    const float* __restrict__ z, const float* __restrict__ Wz,
    const float* __restrict__ gama,
    const float* __restrict__ q1b, const float* __restrict__ k1b,
    const float* __restrict__ v1b, const float* __restrict__ qpb,
    const float* __restrict__ kpb, const float* __restrict__ q2nb,
    const float* __restrict__ k2nb, const float* __restrict__ v3b,
    float* __restrict__ o1acc, float* __restrict__ o2acc,
    float* __restrict__ o3acc) {
    __shared__ float s_wz[DZ * 16];    // Wz padded to 16 head columns
    __shared__ __align__(16) float s_q1[16 * 128];
    __shared__ __align__(16) float s_qp[16 * 96];
    __shared__ __align__(16) float s_q2n[16 * 8];
    __shared__ float s_wgt[8];
    __shared__ __align__(16) float s_k1[16 * 128];
    __shared__ __align__(16) float s_v1[16 * 128];
    __shared__ __align__(16) float s_kp[16 * 96];
    __shared__ __align__(16) float s_k2n[16 * 8];
    __shared__ __align__(16) float s_v3[16 * 144];
    __shared__ float s_att[256 * 8];   // [i_loc][j_loc][h]

    const int t = threadIdx.x;
    const int wave = t >> 5;
    const int lane = t & 31;
    const int lm = lane & 15;   // N index inside a WMMA tile
    const int hs = lane >> 4;   // which K half this lane carries
    const int i0 = blockIdx.x * 16;
    const int j0base = blockIdx.y * 64;

    // i-side staging via async DMA (ASYNCcnt)
    for (int idx = t; idx < (16 * 128) / 4; idx += 256)
        async_cp16(&s_q1[idx * 4], &q1b[i0 * 128 + idx * 4]);
    for (int idx = t; idx < (16 * 96) / 4; idx += 256)
        async_cp16(&s_qp[idx * 4], &qpb[i0 * 96 + idx * 4]);
    if (t < 32) async_cp16(&s_q2n[t * 4], &q2nb[i0 * 8 + t * 4]);

    // Wz needs head padding -> manual stage
    for (int idx = t; idx < DZ * 16; idx += 256) {
        int h = idx & 15;
        s_wz[idx] = (h < 8) ? Wz[(idx >> 4) * 8 + h] : 0.f;
    }
    if (t < 8) s_wgt[t] = log1pf(expf(gama[t])) * WGT_SCALE;

    // Per-wave accumulators: 2 i-rows x 8 dz-tiles of 16x16 f32 (o1),
    // plus per-thread o2 (h dim) and o3 (9 columns of 144).
    v8f acc1[2][8];
#pragma unroll
    for (int ii = 0; ii < 2; ++ii)
#pragma unroll
        for (int nt = 0; nt < 8; ++nt) acc1[ii][nt] = vzero8();
    float acc2[8];
#pragma unroll
    for (int h = 0; h < 8; ++h) acc2[h] = 0.f;
    float acc3[9];
#pragma unroll
    for (int cc = 0; cc < 9; ++cc) acc3[cc] = 0.f;

    wait_async0();  // i-side staging complete

    for (int jc = 0; jc < 64; jc += 16) {
        const int j0 = j0base + jc;
        __syncthreads();  // previous chunk consumers done

        // ---- k-side staging via async DMA; overlaps step1 WMMA below ----
        for (int idx = t; idx < (16 * 128) / 4; idx += 256) {
            async_cp16(&s_k1[idx * 4], &k1b[j0 * 128 + idx * 4]);
            async_cp16(&s_v1[idx * 4], &v1b[j0 * 128 + idx * 4]);
        }
        for (int idx = t; idx < (16 * 96) / 4; idx += 256)
            async_cp16(&s_kp[idx * 4], &kpb[j0 * 96 + idx * 4]);
        for (int idx = t; idx < (16 * 144) / 4; idx += 256)
            async_cp16(&s_v3[idx * 4], &v3b[j0 * 144 + idx * 4]);
        if (t < 32) async_cp16(&s_k2n[t * 4], &k2nb[j0 * 8 + t * 4]);

        // ---- step1: bias_z tiles via WMMA ----
        // Wave handles i_loc = 2*wave, 2*wave+1.  Tile M = 16 j-rows, N = 16
        // (8 heads + padding), K = 128 dz in 32 steps of 4.
        {
            const float* zr0 = z + ((size_t)(i0 + 2 * wave) * LL + (j0 + lm)) * DZ;
            const float* zr1 = zr0 + (size_t)LL * DZ;
            v8f c0 = vzero8();
            v8f c1 = vzero8();
#pragma unroll 4
            for (int k = 0; k < DZ; k += 4) {
                const int k0 = k + 2 * hs;
                v2f b;
                b.x = s_wz[k0 * 16 + lm];
                b.y = s_wz[(k0 + 1) * 16 + lm];
                v2f a0 = *(const v2f*)(zr0 + k0);
                v2f a1 = *(const v2f*)(zr1 + k0);
                c0 = WMMA_F32X4(a0, b, c0);
                c1 = WMMA_F32X4(a1, b, c1);
            }
            if (lm < 8) {
                const int il0 = 2 * wave, il1 = il0 + 1;
#pragma unroll
                for (int r = 0; r < 8; ++r) {
                    int jl = r + 8 * hs;  // C-layout: M = r + 8*(lane>=16)
                    s_att[(il0 * 16 + jl) * 8 + lm] = c0[r];
                    s_att[(il1 * 16 + jl) * 8 + lm] = c1[r];
                }
            }
        }
        wait_async0();   // retire k-side DMA (overlapped with the WMMA loop)
        __syncthreads();

        // ---- step2: attmap + distance term (thread owns one (i,j) pair) ----
        {
            const int il = t >> 4, jl = t & 15;
            const float* q1r = &s_q1[il * 128];
            const float* k1r = &s_k1[jl * 128];
            const float* qpr = &s_qp[il * 96];
            const float* kpr = &s_kp[jl * 96];
#pragma unroll
            for (int h = 0; h < 8; ++h) {
                float dq = 0.f;
#pragma unroll
                for (int c = 0; c < 16; ++c) dq = fmaf(q1r[h * 16 + c], k1r[h * 16 + c], dq);
                float cr = 0.f;
#pragma unroll
                for (int a = 0; a < 12; ++a) cr = fmaf(qpr[h * 12 + a], kpr[h * 12 + a], cr);
                float dis = s_q2n[il * 8 + h] + s_k2n[jl * 8 + h] - 2.f * cr;
                s_att[t * 8 + h] += dq * C_SQU - s_wgt[h] * dis;
            }
        }
        __syncthreads();

        // ---- step3a: o1 += att * z via WMMA ----
        // A: M = head (padded to 16), K = 4 j's.  B: K = 4 j's, N = 16 dz.
#pragma unroll
        for (int ii = 0; ii < 2; ++ii) {
            const int il = 2 * wave + ii;
            const float* zi = z + ((size_t)(i0 + il) * LL + j0) * DZ;
#pragma unroll
            for (int k = 0; k < 16; k += 4) {
                const int k0 = k + 2 * hs;
                v2f a;
                a.x = (lm < 8) ? s_att[(il * 16 + k0) * 8 + lm] : 0.f;
                a.y = (lm < 8) ? s_att[(il * 16 + k0 + 1) * 8 + lm] : 0.f;
#pragma unroll
                for (int nt = 0; nt < 8; ++nt) {
                    v2f b;
                    b.x = zi[(size_t)k0 * DZ + nt * 16 + lm];
                    b.y = zi[(size_t)(k0 + 1) * DZ + nt * 16 + lm];
                    acc1[ii][nt] = WMMA_F32X4(a, b, acc1[ii][nt]);
                }
            }
        }

        // ---- step3b: o2 (B depends on h -> VALU) ----
        {
            const int il = t >> 4, c = t & 15;
#pragma unroll
            for (int h = 0; h < 8; ++h) {
                float sa = 0.f;
#pragma unroll
                for (int j = 0; j < 16; ++j)
                    sa = fmaf(s_att[(il * 16 + j) * 8 + h], s_v1[(j * 8 + h) * 16 + c], sa);
                acc2[h] += sa;
            }
        }
        // ---- step3c: o3 ----
        {
            const int il = t >> 4;
#pragma unroll
            for (int cc = 0; cc < 9; ++cc) {
                int col = (t & 15) * 9 + cc;
                int h = col / 18;
                int pa = col - h * 18;
                float sa = 0.f;
#pragma unroll
                for (int j = 0; j < 16; ++j)
                    sa = fmaf(s_att[(il * 16 + j) * 8 + h], s_v3[j * 144 + h * 18 + pa], sa);
                acc3[cc] += sa;
            }
        }

        // Prefetch next chunk's z rows (CDNA5 global_prefetch_b8)
        if (jc + 16 < 64) {
            const float* pf = z + ((size_t)(i0 + (t >> 4)) * LL + (j0 + 16 + (t & 15))) * DZ;
            __builtin_prefetch(pf, 0, 1);
        }
    }

    // ---- flush partials (16 j-segments race on the same outputs) ----
    if (hs == 0) {  // lanes carrying valid head rows h = r (0..7)
#pragma unroll
        for (int ii = 0; ii < 2; ++ii) {
            const int gi = i0 + 2 * wave + ii;
#pragma unroll
            for (int nt = 0; nt < 8; ++nt)
#pragma unroll
                for (int r = 0; r < 8; ++r)
                    atomicAdd(&o1acc[((size_t)gi * 8 + r) * 128 + nt * 16 + lm],
                              acc1[ii][nt][r]);
        }
    }
    {
        const int gi = i0 + (t >> 4);
        const int c = t & 15;
#pragma unroll
        for (int h = 0; h < 8; ++h)
            atomicAdd(&o2acc[((size_t)gi * 8 + h) * 16 + c], acc2[h]);
#pragma unroll
        for (int cc = 0; cc < 9; ++cc)
            atomicAdd(&o3acc[(size_t)gi * 144 + (t & 15) * 9 + cc], acc3[cc]);
    }
}

// ---------------------------------------------------------------------------
// Inverse frame transform of o3 + pack ovec = [o1|o2|o3'] per row (1296 cols)
// ---------------------------------------------------------------------------
__global__ __launch_bounds__(128) void ipa_gather_kernel(
    const float* __restrict__ o1acc, const float* __restrict__ o2acc,
    const float* __restrict__ o3acc, const float* __restrict__ rot,
    const float* __restrict__ trans, float* __restrict__ ovec) {
    const int i = blockIdx.x;
    const int t = threadIdx.x;
    float* ov = ovec + (size_t)i * 1296;
    for (int idx = t; idx < 1024; idx += 128) ov[idx] = o1acc[(size_t)i * 1024 + idx];
    ov[1024 + t] = o2acc[(size_t)i * 128 + t];
    if (t < 48) {
        int base = t * 3;
        float m0 = o3acc[(size_t)i * 144 + base + 0] - trans[i * 3 + 0];
        float m1 = o3acc[(size_t)i * 144 + base + 1] - trans[i * 3 + 1];
        float m2 = o3acc[(size_t)i * 144 + base + 2] - trans[i * 3 + 2];
#pragma unroll
        for (int d = 0; d < 3; ++d)
            ov[1152 + base + d] = m0 * rot[i * 9 + d * 3 + 0] +
                                  m1 * rot[i * 9 + d * 3 + 1] +
                                  m2 * rot[i * 9 + d * 3 + 2];
    }
}

// ---------------------------------------------------------------------------
// Final GEMM: out[1024,384] = ovec[1024,1296] @ Wcat + bsum, f32 WMMA.
// One 16-row M-tile per block; each wave owns 3 of the 24 N-tiles.
// ---------------------------------------------------------------------------
__global__ __launch_bounds__(256) void ipa_out_gemm_kernel(
    const float* __restrict__ ovec, const float* __restrict__ Wcat,
    const float* __restrict__ bsum, float* __restrict__ out) {
    const int t = threadIdx.x;
    const int wave = t >> 5;
    const int lane = t & 31;
    const int lm = lane & 15;
    const int hs = lane >> 4;
    const int m0 = blockIdx.x * 16;
    const float* Arow = ovec + (size_t)(m0 + lm) * 1296;

    v8f acc[3];
#pragma unroll
    for (int q = 0; q < 3; ++q) acc[q] = vzero8();

    for (int k = 0; k < 1296; k += 4) {
        const int k0 = k + 2 * hs;
        v2f a = *(const v2f*)(Arow + k0);
#pragma unroll
        for (int q = 0; q < 3; ++q) {
            const int nb = (wave * 3 + q) * 16;
            v2f b;
            b.x = Wcat[(size_t)k0 * 384 + nb + lm];
            b.y = Wcat[(size_t)(k0 + 1) * 384 + nb + lm];
            acc[q] = WMMA_F32X4(a, b, acc[q]);
        }
    }
#pragma unroll
    for (int q = 0; q < 3; ++q) {
        const int nb = (wave * 3 + q) * 16;
        const int n = nb + lm;
        const float bb = bsum[n];
#pragma unroll
        for (int r = 0; r < 8; ++r)
            out[(size_t)(m0 + r + 8 * hs) * 384 + n] = acc[q][r] + bb;
    }
}

// ---------------------------------------------------------------------------
// Launch
// ---------------------------------------------------------------------------
extern "C" void kernel_launch(void* const* d_in, const int* in_sizes, int n_in,
                              void* d_out, int out_size, void* d_ws, size_t ws_size,
                              hipStream_t stream) {
    const float* s     = (const float*)d_in[0];
    const float* z     = (const float*)d_in[1];
    const float* rot   = (const float*)d_in[2];
    const float* trans = (const float*)d_in[3];
    const float* Wq1   = (const float*)d_in[4];
    const float* Wk1   = (const float*)d_in[5];
    const float* Wv1   = (const float*)d_in[6];
    const float* Wq2   = (const float*)d_in[7];
    const float* Wk2   = (const float*)d_in[8];
    const float* Wv3   = (const float*)d_in[9];
    const float* Wz    = (const float*)d_in[10];
    const float* W1    = (const float*)d_in[11];
    const float* b1    = (const float*)d_in[12];
    const float* W2    = (const float*)d_in[13];
    const float* b2    = (const float*)d_in[14];
    const float* W3    = (const float*)d_in[15];
    const float* b3    = (const float*)d_in[16];
    const float* gama  = (const float*)d_in[17];

    float* ws = (float*)d_ws;
    float* Q1B  = ws;                 // 131072
    float* K1B  = Q1B + 131072;       // 131072
    float* V1B  = K1B + 131072;       // 131072
    float* QPB  = V1B + 131072;       // 98304
    float* KPB  = QPB + 98304;        // 98304
    float* Q2N  = KPB + 98304;        // 8192
    float* K2N  = Q2N + 8192;         // 8192
    float* V3B  = K2N + 8192;         // 147456
    float* O1A  = V3B + 147456;       // 1048576  -- contiguous atomic region
    float* O2A  = O1A + 1048576;      // 131072
    float* O3A  = O2A + 131072;       // 147456
    float* OVEC = O3A + 147456;       // 1327104
    float* WCAT = OVEC + 1327104;     // 497664
    float* BSUM = WCAT + 497664;      // 384

    // zero the atomic accumulators (o1|o2|o3 are contiguous)
    {
        const int nz = 1048576 + 131072 + 147456;
        ipa_zero_kernel<<<(nz + 255) / 256, 256, 0, stream>>>(O1A, nz);
    }

    ipa_proj_kernel<<<LL, 128, 0, stream>>>(s, rot, trans, Wq1, Wk1, Wv1, Wq2, Wk2,
                                            Wv3, Q1B, K1B, V1B, QPB, KPB, Q2N, K2N,
                                            V3B);

    ipa_wcat_kernel<<<(1296 * 384 + 255) / 256, 256, 0, stream>>>(W1, b1, W2, b2, W3,
                                                                  b3, WCAT, BSUM);

    ipa_att_kernel<<<dim3(64, 16), 256, 0, stream>>>(z, Wz, gama, Q1B, K1B, V1B, QPB,
                                                     KPB, Q2N, K2N, V3B, O1A, O2A,
                                                     O3A);

    ipa_gather_kernel<<<LL, 128, 0, stream>>>(O1A, O2A, O3A, rot, trans, OVEC);

    ipa_out_gemm_kernel<<<64, 256, 0, stream>>>(OVEC, WCAT, BSUM, (float*)d_out);
}